// F2_BNLSTM_Filterbank_89232240541863
// MI455X (gfx1250) — compile-verified
//
#include <hip/hip_runtime.h>
#include <hip/hip_bf16.h>

// Problem constants (from reference)
constexpr int Bn = 8192;   // batch
constexpr int Sn = 29;     // seq len
constexpr int Fn = 129;    // stft features
constexpr int Hn = 32;     // hidden == nfilt
constexpr int G4 = 128;    // 4*H
constexpr int N2 = 256;    // both directions' 4H
constexpr float EPSc = 1e-5f;

typedef __attribute__((ext_vector_type(2))) float v2f;
typedef __attribute__((ext_vector_type(8))) float v8f;

__device__ __forceinline__ float sigmoidf_(float x) { return 1.0f / (1.0f + __expf(-x)); }

// ---------------------------------------------------------------------------
// Init: zero all BN stat accumulators, broadcast h0/c0 into current state.
// ---------------------------------------------------------------------------
__global__ void init_kernel(float* __restrict__ stats, int nstats,
                            const float* __restrict__ h0, const float* __restrict__ c0,
                            float* __restrict__ h_cur, float* __restrict__ c_cur) {
  int i = blockIdx.x * 256 + threadIdx.x;
  if (i < nstats) stats[i] = 0.0f;
  if (i < 2 * Bn * Hn) {
    int h = i & 31;
    int dir = i >> 18;            // 2*8192*32 = 2^19
    h_cur[i] = h0[dir * Hn + h];
    c_cur[i] = c0[dir * Hn + h];
  }
}

// ---------------------------------------------------------------------------
// C = filters @ Wih  (both directions), padded to 132 rows (rows >=129 zero)
// Cmat layout: [132][256] row-major; cols 0..127 fwd, 128..255 bwd.
// ---------------------------------------------------------------------------
__global__ void prep_C(const float* __restrict__ filters,
                       const float* __restrict__ WihF, const float* __restrict__ WihB,
                       float* __restrict__ Cmat) {
  int idx = blockIdx.x * 256 + threadIdx.x;
  if (idx >= 132 * N2) return;
  int f = idx / N2, n = idx % N2;
  float acc = 0.0f;
  if (f < Fn) {
    const float* W = (n < G4) ? WihF : WihB;
    int nn = n & (G4 - 1);
    #pragma unroll
    for (int k = 0; k < Hn; ++k) acc += filters[f * Hn + k] * W[k * G4 + nn];
  }
  Cmat[idx] = acc;
}

// ---------------------------------------------------------------------------
// XW[s][b][n] = (x_stft[b,s,:] @ Cmat)[n]  via V_WMMA_F32_16X16X4_F32.
// Block = 128 threads (4 waves). Wave w covers N = w*64 .. w*64+63 (4 accs).
// Also accumulates per-(s,n) column sum / sumsq in f32 for BN.
// ---------------------------------------------------------------------------
__global__ __launch_bounds__(128) void xw_gemm(const float* __restrict__ x,
                                               const float* __restrict__ Cmat,
                                               __hip_bfloat16* __restrict__ XW,
                                               float* __restrict__ sxw_sum,
                                               float* __restrict__ sxw_sq) {
  __shared__ float Atile[16 * 132];
  int bx = blockIdx.x;
  int s  = bx / (Bn / 16);
  int m0 = (bx % (Bn / 16)) * 16;
  int tid = threadIdx.x;

  // Cooperative, coalesced A-tile load (zero-padded K 129..131)
  for (int idx = tid; idx < 16 * 132; idx += 128) {
    int r = idx / 132, f = idx % 132;
    float v = 0.0f;
    if (f < Fn) v = x[(size_t)(m0 + r) * (Sn * Fn) + (size_t)s * Fn + f];
    Atile[idx] = v;
  }
  __syncthreads();

  int wave = tid >> 5, lane = tid & 31;
  int nbase = wave * 64;
  int mrow  = lane & 15;
  int khalf = (lane >> 4) * 2;

  v8f acc[4];
  #pragma unroll
  for (int j = 0; j < 4; ++j)
    #pragma unroll
    for (int e = 0; e < 8; ++e) acc[j][e] = 0.0f;

  for (int kk = 0; kk < 132; kk += 4) {     // 33 K-iterations
    v2f a;
    a.x = Atile[mrow * 132 + kk + khalf];
    a.y = Atile[mrow * 132 + kk + khalf + 1];
    #pragma unroll
    for (int j = 0; j < 4; ++j) {
      int n = nbase + j * 16 + mrow;
      v2f bf;
      bf.x = Cmat[(size_t)(kk + khalf) * N2 + n];
      bf.y = Cmat[(size_t)(kk + khalf + 1) * N2 + n];
      acc[j] = __builtin_amdgcn_wmma_f32_16x16x4_f32(
          false, a, false, bf, (short)0, acc[j], false, false);
    }
  }

  // Store D (bf16) + per-column BN stats (f32)
  #pragma unroll
  for (int j = 0; j < 4; ++j) {
    int n = nbase + j * 16 + mrow;
    float csum = 0.0f, csq = 0.0f;
    #pragma unroll
    for (int r = 0; r < 8; ++r) {
      int row = m0 + r + ((lane >> 4) << 3);
      float v = acc[j][r];
      XW[((size_t)s * Bn + row) * N2 + n] = __float2bfloat16(v);
      csum += v; csq += v * v;
    }
    csum += __shfl_xor(csum, 16);
    csq  += __shfl_xor(csq, 16);
    if (lane < 16) {
      atomicAdd(&sxw_sum[s * N2 + n], csum);
      atomicAdd(&sxw_sq [s * N2 + n], csq);
    }
  }
}

// ---------------------------------------------------------------------------
// hW = h_cur @ Whh for both directions (K=32). Waves 0,1 -> fwd; 2,3 -> bwd.
// ---------------------------------------------------------------------------
__global__ __launch_bounds__(128) void hw_gemm(const float* __restrict__ h_cur,
                                               const float* __restrict__ WhhF,
                                               const float* __restrict__ WhhB,
                                               float* __restrict__ hwbuf,
                                               float* __restrict__ shw_sum,
                                               float* __restrict__ shw_sq, int t) {
  int m0 = blockIdx.x * 16;
  int tid = threadIdx.x, wave = tid >> 5, lane = tid & 31;
  int dir = wave >> 1;
  int nbase = (wave & 1) * 64;
  const float* W = dir ? WhhB : WhhF;
  const float* hb = h_cur + (size_t)dir * Bn * Hn;
  int mrow = lane & 15, khalf = (lane >> 4) * 2;

  v8f acc[4];
  #pragma unroll
  for (int j = 0; j < 4; ++j)
    #pragma unroll
    for (int e = 0; e < 8; ++e) acc[j][e] = 0.0f;

  #pragma unroll
  for (int kk = 0; kk < Hn; kk += 4) {      // 8 K-iterations
    v2f a;
    a.x = hb[(size_t)(m0 + mrow) * Hn + kk + khalf];
    a.y = hb[(size_t)(m0 + mrow) * Hn + kk + khalf + 1];
    #pragma unroll
    for (int j = 0; j < 4; ++j) {
      int n = nbase + j * 16 + mrow;
      v2f bf;
      bf.x = W[(kk + khalf) * G4 + n];
      bf.y = W[(kk + khalf + 1) * G4 + n];
      acc[j] = __builtin_amdgcn_wmma_f32_16x16x4_f32(
          false, a, false, bf, (short)0, acc[j], false, false);
    }
  }

  #pragma unroll
  for (int j = 0; j < 4; ++j) {
    int n = nbase + j * 16 + mrow;
    float csum = 0.0f, csq = 0.0f;
    #pragma unroll
    for (int r = 0; r < 8; ++r) {
      int row = m0 + r + ((lane >> 4) << 3);
      float v = acc[j][r];
      hwbuf[((size_t)dir * Bn + row) * G4 + n] = v;
      csum += v; csq += v * v;
    }
    csum += __shfl_xor(csum, 16);
    csq  += __shfl_xor(csq, 16);
    if (lane < 16) {
      atomicAdd(&shw_sum[(dir * Sn + t) * G4 + n], csum);
      atomicAdd(&shw_sq [(dir * Sn + t) * G4 + n], csq);
    }
  }
}

// ---------------------------------------------------------------------------
// Gates: hid = BN(xW) + BN(hW) + b -> new_c, sigmoid(o); accumulate c stats.
// grid = (B/8, 2[dir]); block 256 = 8 rows x 32 cols.
// ---------------------------------------------------------------------------
__global__ __launch_bounds__(256) void gates_kernel(
    const __hip_bfloat16* __restrict__ XW, const float* __restrict__ hwbuf,
    const float* __restrict__ sxw_sum, const float* __restrict__ sxw_sq,
    const float* __restrict__ shw_sum, const float* __restrict__ shw_sq,
    const float* __restrict__ bF, const float* __restrict__ bB,
    float* __restrict__ c_cur, float* __restrict__ sig_o,
    float* __restrict__ sc_sum, float* __restrict__ sc_sq, int t) {
  int dir = blockIdx.y;
  int tid = threadIdx.x;
  int h = tid & 31;
  int b = blockIdx.x * 8 + (tid >> 5);
  int s = dir ? (Sn - 1 - t) : t;
  const float* bvec = dir ? bB : bF;
  const float invB = 1.0f / (float)Bn;

  float hid[4];
  #pragma unroll
  for (int q = 0; q < 4; ++q) {
    int col = q * Hn + h;
    int nx = dir * G4 + col;
    float xv = __bfloat162float(XW[((size_t)s * Bn + b) * N2 + nx]);
    float mx = sxw_sum[s * N2 + nx] * invB;
    float vx = sxw_sq [s * N2 + nx] * invB - mx * mx;
    float xn = (xv - mx) * rsqrtf(fmaxf(vx, 0.0f) + EPSc);
    float hv = hwbuf[((size_t)dir * Bn + b) * G4 + col];
    float mh = shw_sum[(dir * Sn + t) * G4 + col] * invB;
    float vh = shw_sq [(dir * Sn + t) * G4 + col] * invB - mh * mh;
    float hn = (hv - mh) * rsqrtf(fmaxf(vh, 0.0f) + EPSc);
    hid[q] = xn + hn + bvec[col];
  }
  float ig = sigmoidf_(hid[0]);
  float fg = sigmoidf_(hid[1]);
  float og = sigmoidf_(hid[2]);
  float gg = tanhf(hid[3]);

  size_t ci = ((size_t)dir * Bn + b) * Hn + h;
  float nc = fg * c_cur[ci] + ig * gg;
  c_cur[ci] = nc;
  sig_o[ci] = og;

  __shared__ float ssum[256];
  __shared__ float ssq[256];
  ssum[tid] = nc;
  ssq[tid]  = nc * nc;
  __syncthreads();
  if (tid < 32) {
    float s1 = 0.0f, s2 = 0.0f;
    #pragma unroll
    for (int r = 0; r < 8; ++r) { s1 += ssum[h + r * 32]; s2 += ssq[h + r * 32]; }
    atomicAdd(&sc_sum[(dir * Sn + t) * Hn + h], s1);
    atomicAdd(&sc_sq [(dir * Sn + t) * Hn + h], s2);
  }
}

// ---------------------------------------------------------------------------
// new_h = sigmoid(o) * tanh(BN(new_c)*gc + bc); write h_cur and h_all[s].
// ---------------------------------------------------------------------------
__global__ void newh_kernel(const float* __restrict__ c_cur, const float* __restrict__ sig_o,
                            const float* __restrict__ sc_sum, const float* __restrict__ sc_sq,
                            const float* __restrict__ gcF, const float* __restrict__ bcF,
                            const float* __restrict__ gcB, const float* __restrict__ bcB,
                            float* __restrict__ h_cur, float* __restrict__ h_all, int t) {
  int i = blockIdx.x * 256 + threadIdx.x;
  if (i >= 2 * Bn * Hn) return;
  int h = i & 31;
  int b = (i >> 5) & (Bn - 1);
  int dir = i >> 18;
  const float invB = 1.0f / (float)Bn;
  float mu = sc_sum[(dir * Sn + t) * Hn + h] * invB;
  float var = sc_sq[(dir * Sn + t) * Hn + h] * invB - mu * mu;
  float cn = (c_cur[i] - mu) * rsqrtf(fmaxf(var, 0.0f) + EPSc);
  float g = dir ? gcB[h] : gcF[h];
  float bb = dir ? bcB[h] : bcF[h];
  float hv = sig_o[i] * tanhf(cn * g + bb);
  h_cur[i] = hv;
  int s = dir ? (Sn - 1 - t) : t;
  h_all[(((size_t)dir * Sn + s) * Bn + b) * Hn + h] = hv;
}

// ---------------------------------------------------------------------------
// Attention scores: score[b,s] = ae . tanh(Wa @ hid[b,s] + ba)
// ---------------------------------------------------------------------------
__global__ __launch_bounds__(256) void score_kernel(const float* __restrict__ h_all,
                                                    const float* __restrict__ Wa,
                                                    const float* __restrict__ ba,
                                                    const float* __restrict__ ae,
                                                    float* __restrict__ scores) {
  __shared__ float sWa[64 * 64];
  __shared__ float sba[64];
  __shared__ float sae[64];
  int tid = threadIdx.x;
  for (int idx = tid; idx < 64 * 64; idx += 256) sWa[idx] = Wa[idx];
  if (tid < 64) { sba[tid] = ba[tid]; sae[tid] = ae[tid]; }
  __syncthreads();

  int p = blockIdx.x * 256 + tid;
  if (p >= Bn * Sn) return;
  int b = p / Sn, s = p % Sn;
  float v[64];
  const float* hf = &h_all[(((size_t)0 * Sn + s) * Bn + b) * Hn];
  const float* hb = &h_all[(((size_t)1 * Sn + s) * Bn + b) * Hn];
  #pragma unroll
  for (int k = 0; k < 32; ++k) { v[k] = hf[k]; v[32 + k] = hb[k]; }
  float sc = 0.0f;
  for (int j = 0; j < 64; ++j) {
    float a = sba[j];
    #pragma unroll
    for (int k = 0; k < 64; ++k) a += sWa[j * 64 + k] * v[k];
    sc += sae[j] * tanhf(a);
  }
  scores[b * Sn + s] = sc;
}

// ---------------------------------------------------------------------------
// Softmax over S + weighted sum: out[b, 0:64] (fwd cols 0..31, bwd 32..63)
// block 256 = 4 b x 64 n.
// ---------------------------------------------------------------------------
__global__ __launch_bounds__(256) void attn_out(const float* __restrict__ h_all,
                                                const float* __restrict__ scores,
                                                float* __restrict__ out) {
  int tid = threadIdx.x;
  int n = tid & 63;
  int b = blockIdx.x * 4 + (tid >> 6);
  float sc[Sn];
  float m = -1e30f;
  #pragma unroll
  for (int s = 0; s < Sn; ++s) { sc[s] = scores[b * Sn + s]; m = fmaxf(m, sc[s]); }
  float sum = 0.0f;
  #pragma unroll
  for (int s = 0; s < Sn; ++s) { sc[s] = __expf(sc[s] - m); sum += sc[s]; }
  float inv = 1.0f / sum;
  int dir = n >> 5, h = n & 31;
  float acc = 0.0f;
  #pragma unroll
  for (int s = 0; s < Sn; ++s)
    acc += sc[s] * h_all[(((size_t)dir * Sn + s) * Bn + b) * Hn + h];
  out[b * 64 + n] = acc * inv;
}

// ---------------------------------------------------------------------------
// Host-side launcher
// ---------------------------------------------------------------------------
extern "C" void kernel_launch(void* const* d_in, const int* in_sizes, int n_in,
                              void* d_out, int out_size, void* d_ws, size_t ws_size,
                              hipStream_t stream) {
  const float* x       = (const float*)d_in[0];
  const float* filters = (const float*)d_in[1];
  const float* WihF    = (const float*)d_in[2];
  const float* WhhF    = (const float*)d_in[3];
  const float* bF      = (const float*)d_in[4];
  const float* gcF     = (const float*)d_in[5];
  const float* bcF     = (const float*)d_in[6];
  const float* WihB    = (const float*)d_in[7];
  const float* WhhB    = (const float*)d_in[8];
  const float* bB      = (const float*)d_in[9];
  const float* gcB     = (const float*)d_in[10];
  const float* bcB     = (const float*)d_in[11];
  const float* h0      = (const float*)d_in[12];
  const float* c0      = (const float*)d_in[13];
  const float* Wa      = (const float*)d_in[14];
  const float* ba      = (const float*)d_in[15];
  const float* ae      = (const float*)d_in[16];

  // Workspace carve (~198 MB total)
  char* p = (char*)d_ws;
  auto take = [&](size_t bytes) -> char* {
    char* r = p;
    p += (bytes + 255) & ~(size_t)255;
    return r;
  };
  float*          Cmat  = (float*)take((size_t)132 * N2 * sizeof(float));
  __hip_bfloat16* XW    = (__hip_bfloat16*)take((size_t)Sn * Bn * N2 * sizeof(__hip_bfloat16));
  const int NSTATS = 2 * Sn * N2 + 4 * Sn * G4 + 4 * Sn * Hn;   // 33408 floats
  float* stats   = (float*)take((size_t)NSTATS * sizeof(float));
  float* sxw_sum = stats;
  float* sxw_sq  = sxw_sum + Sn * N2;
  float* shw_sum = sxw_sq  + Sn * N2;
  float* shw_sq  = shw_sum + 2 * Sn * G4;
  float* sc_sum  = shw_sq  + 2 * Sn * G4;
  float* sc_sq   = sc_sum  + 2 * Sn * Hn;
  float* hwbuf  = (float*)take((size_t)2 * Bn * G4 * sizeof(float));
  float* sig_o  = (float*)take((size_t)2 * Bn * Hn * sizeof(float));
  float* c_cur  = (float*)take((size_t)2 * Bn * Hn * sizeof(float));
  float* h_cur  = (float*)take((size_t)2 * Bn * Hn * sizeof(float));
  float* h_all  = (float*)take((size_t)2 * Sn * Bn * Hn * sizeof(float));
  float* scores = (float*)take((size_t)Bn * Sn * sizeof(float));
  (void)ws_size; (void)in_sizes; (void)n_in; (void)out_size;

  // Setup
  init_kernel<<<(2 * Bn * Hn + 255) / 256, 256, 0, stream>>>(stats, NSTATS, h0, c0, h_cur, c_cur);
  prep_C<<<(132 * N2 + 255) / 256, 256, 0, stream>>>(filters, WihF, WihB, Cmat);

  // Big fused filterbank+input GEMM (all steps, both directions) with f32 WMMA
  xw_gemm<<<Sn * (Bn / 16), 128, 0, stream>>>(x, Cmat, XW, sxw_sum, sxw_sq);

  // Sequential bidirectional BN-LSTM scan
  for (int t = 0; t < Sn; ++t) {
    hw_gemm<<<Bn / 16, 128, 0, stream>>>(h_cur, WhhF, WhhB, hwbuf, shw_sum, shw_sq, t);
    dim3 gg(Bn / 8, 2);
    gates_kernel<<<gg, 256, 0, stream>>>(XW, hwbuf, sxw_sum, sxw_sq, shw_sum, shw_sq,
                                         bF, bB, c_cur, sig_o, sc_sum, sc_sq, t);
    newh_kernel<<<(2 * Bn * Hn + 255) / 256, 256, 0, stream>>>(
        c_cur, sig_o, sc_sum, sc_sq, gcF, bcF, gcB, bcB, h_cur, h_all, t);
  }

  // Attention aggregator
  score_kernel<<<(Bn * Sn + 255) / 256, 256, 0, stream>>>(h_all, Wa, ba, ae, scores);
  attn_out<<<Bn / 4, 256, 0, stream>>>(h_all, scores, (float*)d_out);
}